// ContourIntegrationLayer_4535485464773
// MI455X (gfx1250) — compile-verified
//
#include <hip/hip_runtime.h>

// Depthwise 3x3 conv (SAME) + residual, NHWC (32,112,112,96) fp32.
// Memory-bound (~13us HBM floor at 23.3 TB/s). Strategy: async-copy 3 input
// rows into LDS (CDNA5 GLOBAL_LOAD_ASYNC_TO_LDS_B128 / ASYNCcnt path), then
// run the stencil out of LDS with 128-bit accesses everywhere.

#define CH     96
#define C4     (CH / 4)          // 24 float4 channel-groups
#define TW     16                // output pixels per tile
#define HWID   (TW + 2)          // 18 halo pixels
#define NROWS  3
#define LDS_F4 (NROWS * HWID * C4)   // 1296 float4 = 20736 B
#define BLOCK  (TW * C4)             // 384 threads = 12 waves (wave32)

#define IMG_H  112
#define IMG_W  112
#define TILES_W (IMG_W / TW)     // 7

__device__ __forceinline__ void fma4(float4& acc, const float4& w, const float4& v) {
    acc.x = fmaf(w.x, v.x, acc.x);
    acc.y = fmaf(w.y, v.y, acc.y);
    acc.z = fmaf(w.z, v.z, acc.z);
    acc.w = fmaf(w.w, v.w, acc.w);
}

__global__ __launch_bounds__(BLOCK) void contour_dwconv3x3_residual(
    const float* __restrict__ x,    // (32,112,112,96)
    const float* __restrict__ kern, // (3,3,1,96)
    float* __restrict__ out)
{
    __shared__ float4 tile[LDS_F4];

    const int tid = threadIdx.x;
    const int bid = blockIdx.x;

    const int tw = bid % TILES_W;
    const int h  = (bid / TILES_W) % IMG_H;
    const int n  =  bid / (TILES_W * IMG_H);
    const int w0 = tw * TW;

    const size_t row_stride = (size_t)IMG_W * CH;       // floats
    const size_t img_stride = (size_t)IMG_H * row_stride;
    const float* xbase = x + (size_t)n * img_stride;

    // ---- Zero-fill only when some halo is out of bounds (SAME padding) ----
    const bool boundary = (h == 0) | (h == IMG_H - 1) | (tw == 0) | (tw == TILES_W - 1);
    if (boundary) {
        for (int i = tid; i < LDS_F4; i += BLOCK)
            tile[i] = make_float4(0.f, 0.f, 0.f, 0.f);
        __syncthreads();   // zeros visible before async writes land
    }

    // ---- Stage 3 halo rows into LDS via async-to-LDS b128 (ASYNCcnt) ----
    for (int i = tid; i < LDS_F4; i += BLOCK) {
        int r   = i / (HWID * C4);
        int rem = i - r * (HWID * C4);
        int p   = rem / C4;
        int c4g = rem - p * C4;
        int hg  = h + r - 1;
        int wg  = w0 + p - 1;
        if ((unsigned)hg < (unsigned)IMG_H && (unsigned)wg < (unsigned)IMG_W) {
            const float* gp = xbase + (size_t)hg * row_stride + (size_t)wg * CH + c4g * 4;
            unsigned long long gaddr = (unsigned long long)(uintptr_t)gp;
            // generic pointer to __shared__ has the LDS byte offset in its low 32 bits
            unsigned lds_addr = (unsigned)(uintptr_t)(&tile[i]);
            asm volatile("global_load_async_to_lds_b128 %0, %1, off"
                         :: "v"(lds_addr), "v"(gaddr)
                         : "memory");
        }
    }

    // ---- Load per-channel weights while the async copies are in flight ----
    const int c4g = tid % C4;          // channel group owned by this thread
    const int wl  = tid / C4;          // local output pixel owned by this thread
    const float* kp = kern + c4g * 4;  // (3,3,1,96) -> tap t at kp + t*96
    float4 kw[9];
#pragma unroll
    for (int t = 0; t < 9; ++t)
        kw[t] = *(const float4*)(kp + t * CH);

    // ---- Drain this wave's async copies, then sync all waves ----
    asm volatile("s_wait_asynccnt 0x0" ::: "memory");
    __syncthreads();

    // ---- 3x3 stencil + residual out of LDS ----
    float4 acc = tile[(1 * HWID + (wl + 1)) * C4 + c4g];   // residual: + x
#pragma unroll
    for (int r = 0; r < 3; ++r) {
#pragma unroll
        for (int dw = 0; dw < 3; ++dw) {
            float4 v = tile[(r * HWID + wl + dw) * C4 + c4g];
            fma4(acc, kw[r * 3 + dw], v);
        }
    }

    float* op = out + (size_t)n * img_stride + (size_t)h * row_stride
                    + (size_t)(w0 + wl) * CH + c4g * 4;
    *(float4*)op = acc;
}

extern "C" void kernel_launch(void* const* d_in, const int* in_sizes, int n_in,
                              void* d_out, int out_size, void* d_ws, size_t ws_size,
                              hipStream_t stream) {
    const float* x    = (const float*)d_in[0];
    const float* kern = (const float*)d_in[1];
    float* out        = (float*)d_out;

    const int nblocks = 32 * IMG_H * TILES_W;   // one block per (n, h, w-tile)
    contour_dwconv3x3_residual<<<nblocks, BLOCK, 0, stream>>>(x, kern, out);
}